// AttentionHeadCheb_3401614098845
// MI455X (gfx1250) — compile-verified
//
#include <hip/hip_runtime.h>
#include <hip/hip_bf16.h>

#define N_NODES 50000
#define N_EDGES 800000
#define IN_DIM 128
#define OUT_DIM 64
#define K_KERNELS 2
#define S_ATTEN 2
#define CAT_DIM (IN_DIM + OUT_DIM)   // 192

typedef __attribute__((ext_vector_type(2))) float v2f;
typedef __attribute__((ext_vector_type(8))) float v8f;

// ---- monotonic float<->uint encoding for atomicMax-based segment max ----
__device__ __forceinline__ unsigned encf(float f) {
  unsigned u = __float_as_uint(f);
  return (u & 0x80000000u) ? ~u : (u | 0x80000000u);
}
__device__ __forceinline__ float decf(unsigned e) {
  unsigned u = (e & 0x80000000u) ? (e ^ 0x80000000u) : ~e;
  return __uint_as_float(u);
}

// ---------------- zero scratch ----------------
__global__ void k_zero(float* __restrict__ p, int n) {
  int i = blockIdx.x * blockDim.x + threadIdx.x;
  if (i < n) p[i] = 0.0f;
}

// ---------------- wx[k] = x @ W_transform[k]  (fp32 WMMA) ----------------
// grid: N/16 blocks, 256 threads = 8 waves: wave -> (k in 0..1, colTile in 0..3)
__global__ void __launch_bounds__(256) k_wx(const float* __restrict__ x,
                                            const float* __restrict__ Wt,
                                            float* __restrict__ wx) {
  const int wave = threadIdx.x >> 5;
  const int lane = threadIdx.x & 31;
  const int k = wave >> 2;
  const int col0 = (wave & 3) * 16;
  const int row0 = blockIdx.x * 16;
  const int m    = lane & 15;     // row (A) / col (B,C)
  const int half = lane >> 4;     // 0 or 1
  const int kOff = half * 2;

  const float* __restrict__ W  = Wt + k * (IN_DIM * OUT_DIM);
  const float* __restrict__ xr = x + (size_t)(row0 + m) * IN_DIM;

  v8f c = {};
  for (int kk = 0; kk < IN_DIM; kk += 4) {
    v2f a, b;
    a.x = xr[kk + kOff];
    a.y = xr[kk + kOff + 1];
    b.x = W[(kk + kOff)     * OUT_DIM + col0 + m];
    b.y = W[(kk + kOff + 1) * OUT_DIM + col0 + m];
    c = __builtin_amdgcn_wmma_f32_16x16x4_f32(false, a, false, b, (short)0, c,
                                              false, false);
  }
  float* __restrict__ o = wx + (size_t)k * N_NODES * OUT_DIM;
#pragma unroll
  for (int g = 0; g < 8; ++g)
    o[(size_t)(row0 + g + half * 8) * OUT_DIM + col0 + m] = c[g];
}

// ---------------- al/ar projections (N x 3 per kernel) ----------------
__global__ void k_alar(const float* __restrict__ wx,
                       const float* __restrict__ wl,
                       const float* __restrict__ wr,
                       float* __restrict__ al, float* __restrict__ ar) {
  int tid = blockIdx.x * blockDim.x + threadIdx.x;
  if (tid >= K_KERNELS * N_NODES) return;
  int k = tid / N_NODES;
  int n = tid - k * N_NODES;
  const float* row = wx + (size_t)(k * N_NODES + n) * OUT_DIM;
#pragma unroll
  for (int i = 0; i <= S_ATTEN; ++i) {
    const float* Lw = wl + (size_t)(k * (S_ATTEN + 1) + i) * OUT_DIM;
    const float* Rw = wr + (size_t)(k * (S_ATTEN + 1) + i) * OUT_DIM;
    float sl = 0.f, sr = 0.f;
    for (int d = 0; d < OUT_DIM; ++d) {
      float v = row[d];
      sl += v * Lw[d];
      sr += v * Rw[d];
    }
    al[(size_t)(k * N_NODES + n) * 4 + i] = sl;
    ar[(size_t)(k * N_NODES + n) * 4 + i] = sr;
  }
}

// ---------------- edge scores + segment max (encoded atomicMax) ----------------
__global__ void k_score(const long long* __restrict__ er,
                        const long long* __restrict__ ec,
                        const float* __restrict__ al, const float* __restrict__ ar,
                        const float* __restrict__ atten,
                        const float* __restrict__ support,
                        float* __restrict__ sc, unsigned* __restrict__ mEnc) {
  int e = blockIdx.x * blockDim.x + threadIdx.x;
  if (e >= N_EDGES) return;
  int r = (int)er[e];
  int c = (int)ec[e];
  float a0 = atten[e];
  float a1 = atten[N_EDGES + e];
#pragma unroll
  for (int k = 0; k < K_KERNELS; ++k) {
    const float* alr = al + (size_t)(k * N_NODES + r) * 4;
    const float* arc = ar + (size_t)(k * N_NODES + c) * 4;
    float s = (alr[0] + arc[0]) * a0 + (alr[1] + arc[1]) * a1 +
              (alr[2] + arc[2]) * support[(size_t)k * N_EDGES + e];
    sc[(size_t)k * N_EDGES + e] = s;
    atomicMax(&mEnc[k * N_NODES + r], encf(s));
  }
}

// ---------------- exp(s - m) + segment sum ----------------
__global__ void k_exp(const long long* __restrict__ er,
                      const unsigned* __restrict__ mEnc,
                      float* __restrict__ sc, float* __restrict__ denom) {
  int e = blockIdx.x * blockDim.x + threadIdx.x;
  if (e >= N_EDGES) return;
  int r = (int)er[e];
#pragma unroll
  for (int k = 0; k < K_KERNELS; ++k) {
    float m  = decf(mEnc[k * N_NODES + r]);
    float ex = expf(sc[(size_t)k * N_EDGES + e] - m);
    sc[(size_t)k * N_EDGES + e] = ex;
    atomicAdd(&denom[k * N_NODES + r], ex);
  }
}

// ---------------- normalize + scatter SpMM: acc[r] += p_e * wx[col] ----------------
__global__ void k_scatter(const long long* __restrict__ er,
                          const long long* __restrict__ ec,
                          const float* __restrict__ sc,
                          const float* __restrict__ denom,
                          const float* __restrict__ wx,
                          float* __restrict__ acc) {
  int tid = blockIdx.x * blockDim.x + threadIdx.x;
  if (tid >= N_EDGES * OUT_DIM) return;
  int e = tid >> 6;
  int d = tid & (OUT_DIM - 1);
  int r = (int)er[e];
  int c = (int)ec[e];
  float v = 0.f;
#pragma unroll
  for (int k = 0; k < K_KERNELS; ++k) {
    float p = sc[(size_t)k * N_EDGES + e] / denom[k * N_NODES + r];
    v += p * wx[(size_t)(k * N_NODES + c) * OUT_DIM + d];
  }
  atomicAdd(&acc[(size_t)r * OUT_DIM + d], v);
}

// ---------------- residual GEMM: elu([x|acc] @ W_res)  (fp32 WMMA) ----------------
// grid: N/16 blocks, 128 threads = 4 waves (one per 16-col tile)
__global__ void __launch_bounds__(128) k_resid(const float* __restrict__ x,
                                               const float* __restrict__ acc,
                                               const float* __restrict__ Wr,
                                               float* __restrict__ out) {
  const int wave = threadIdx.x >> 5;
  const int lane = threadIdx.x & 31;
  const int col0 = wave * 16;
  const int row0 = blockIdx.x * 16;
  const int m    = lane & 15;
  const int half = lane >> 4;
  const int kOff = half * 2;

  const float* __restrict__ xr = x   + (size_t)(row0 + m) * IN_DIM;
  const float* __restrict__ ac = acc + (size_t)(row0 + m) * OUT_DIM;

  v8f c = {};
  for (int kk = 0; kk < CAT_DIM; kk += 4) {
    int k0 = kk + kOff;           // k0 and k0+1 are on the same side of 128
    v2f a, b;
    if (k0 < IN_DIM) {
      a.x = xr[k0];
      a.y = xr[k0 + 1];
    } else {
      a.x = ac[k0 - IN_DIM];
      a.y = ac[k0 - IN_DIM + 1];
    }
    b.x = Wr[(size_t)k0 * OUT_DIM + col0 + m];
    b.y = Wr[(size_t)(k0 + 1) * OUT_DIM + col0 + m];
    c = __builtin_amdgcn_wmma_f32_16x16x4_f32(false, a, false, b, (short)0, c,
                                              false, false);
  }
#pragma unroll
  for (int g = 0; g < 8; ++g) {
    float v = c[g];
    v = (v > 0.0f) ? v : expm1f(v);   // ELU, alpha = 1
    out[(size_t)(row0 + g + half * 8) * OUT_DIM + col0 + m] = v;
  }
}

extern "C" void kernel_launch(void* const* d_in, const int* in_sizes, int n_in,
                              void* d_out, int out_size, void* d_ws, size_t ws_size,
                              hipStream_t stream) {
  const float*     x       = (const float*)d_in[0];
  const float*     support = (const float*)d_in[1];
  const float*     atten   = (const float*)d_in[2];
  const float*     Wt      = (const float*)d_in[3];
  const float*     wl      = (const float*)d_in[4];
  const float*     wr      = (const float*)d_in[5];
  const float*     Wr      = (const float*)d_in[6];
  const long long* er      = (const long long*)d_in[7];
  const long long* ec      = (const long long*)d_in[8];
  float*           out     = (float*)d_out;

  // workspace layout (floats)
  float* ws    = (float*)d_ws;
  float* wx    = ws;                                              // K*N*64
  float* acc   = wx  + (size_t)K_KERNELS * N_NODES * OUT_DIM;     // N*64
  float* al    = acc + (size_t)N_NODES * OUT_DIM;                 // K*N*4
  float* ar    = al  + (size_t)K_KERNELS * N_NODES * 4;           // K*N*4
  float* denom = ar  + (size_t)K_KERNELS * N_NODES * 4;           // K*N
  unsigned* mEnc = (unsigned*)(denom + (size_t)K_KERNELS * N_NODES); // K*N
  float* sc    = (float*)(mEnc + (size_t)K_KERNELS * N_NODES);    // K*E

  // zero acc..mEnc (contiguous): N*64 + 2*K*N*4 + K*N + K*N
  int zn = N_NODES * OUT_DIM + 2 * K_KERNELS * N_NODES * 4 + 2 * K_KERNELS * N_NODES;
  k_zero<<<(zn + 255) / 256, 256, 0, stream>>>(acc, zn);

  k_wx<<<N_NODES / 16, 256, 0, stream>>>(x, Wt, wx);
  k_alar<<<(K_KERNELS * N_NODES + 255) / 256, 256, 0, stream>>>(wx, wl, wr, al, ar);
  k_score<<<(N_EDGES + 255) / 256, 256, 0, stream>>>(er, ec, al, ar, atten, support, sc, mEnc);
  k_exp<<<(N_EDGES + 255) / 256, 256, 0, stream>>>(er, mEnc, sc, denom);
  k_scatter<<<(N_EDGES * OUT_DIM + 255) / 256, 256, 0, stream>>>(er, ec, sc, denom, wx, acc);
  k_resid<<<N_NODES / 16, 128, 0, stream>>>(x, acc, Wr, out);
}